// GCN_58480274703249
// MI455X (gfx1250) — compile-verified
//
#include <hip/hip_runtime.h>
#include <hip/hip_bf16.h>

// ---------------- problem constants (match reference) ----------------
#define GN  100000      // nodes
#define GE  1600000     // edges
#define GH  128         // feature width (F_IN == H == 128)
#define GEPS 1e-5f

typedef float v2f __attribute__((ext_vector_type(2)));
typedef float v8f __attribute__((ext_vector_type(8)));

// ---------------- degree / normalization ----------------
__global__ void k_deg_init(float* __restrict__ deg) {
    int n = blockIdx.x * blockDim.x + threadIdx.x;
    if (n < GN) deg[n] = 1.0f;                       // self loop
}

__global__ void k_deg_edges(const int* __restrict__ ei, float* __restrict__ deg) {
    int e = blockIdx.x * blockDim.x + threadIdx.x;
    if (e < GE) atomicAdd(&deg[ei[GE + e]], 1.0f);   // col = ei[1][e]
}

__global__ void k_dis(const float* __restrict__ deg, float* __restrict__ dis) {
    int n = blockIdx.x * blockDim.x + threadIdx.x;
    if (n < GN) dis[n] = rsqrtf(fmaxf(deg[n], 1.0f)); // deg >= 1 always (self loop)
}

// ---------------- fp32 WMMA GEMM: hw = (h @ W) * dis[row]; agg = hw (self-loop init) ----
// One wave -> one 16-row strip across all 128 output columns (8 N-tiles).
// A 16x4 f32 layout: lane l holds M=l&15, K-pair base = (l>>4)*2 (VGPR0: K+0, VGPR1: K+1)
// B 4x16 f32 layout (symmetric): lane l holds N=l&15, same K-pair striping.
// C/D 16x16 f32: lane l col = l&15, VGPR v row = v + 8*(l>>4).
//
// W is staged into LDS PRE-PACKED in B-fragment order:
//   sB[((kg*8 + j)*32 + lane)] = { W[kg*4+kk][j*16+nl], W[kg*4+kk+1][j*16+nl] }
// so the inner loop issues one contiguous ds_load_b64 per (kg, j) straight into an
// even-aligned VGPR pair (no v_mov shuffling, no LDS bank conflicts).
__global__ __launch_bounds__(128) void k_gemm_wmma(
    const float* __restrict__ h, const float* __restrict__ W,
    const float* __restrict__ dis,
    float* __restrict__ hw, float* __restrict__ agg)
{
    __shared__ v2f sB[32 * 8 * 32];                  // 64 KB of the 320 KB WGP LDS
    const int tid = threadIdx.x;

    // repack W (row-major [k][n]) into B-fragment order
    for (int idx = tid; idx < 32 * 8 * 32; idx += 128) {
        int l  = idx & 31;
        int j  = (idx >> 5) & 7;
        int kg = idx >> 8;
        int kk = (l >> 4) * 2;
        int nl = l & 15;
        int k  = kg * 4 + kk;
        v2f p;
        p.x = W[(size_t)k       * GH + j * 16 + nl];
        p.y = W[(size_t)(k + 1) * GH + j * 16 + nl];
        sB[idx] = p;
    }
    __syncthreads();

    const int wave  = tid >> 5;
    const int lane  = tid & 31;
    const int strip = blockIdx.x * 4 + wave;
    const int m0    = strip * 16;
    if (m0 >= GN) return;                            // uniform per wave: EXEC stays all-1s

    const int hi = lane >> 4;                        // half-wave select
    const int nl = lane & 15;                        // M (for A) / N (for B)
    const int kk = hi * 2;                           // K-pair base within 4

    v8f acc[8] = {};                                 // 8 col-tiles of 16

    const float* arow = h + (size_t)(m0 + nl) * GH;
    const v2f* sBl = sB + lane;
    for (int kg = 0; kg < 32; ++kg) {
        v2f a = *(const v2f*)&arow[kg * 4 + kk];
        #pragma unroll
        for (int j = 0; j < 8; ++j) {
            v2f b = sBl[(kg * 8 + j) * 32];
            acc[j] = __builtin_amdgcn_wmma_f32_16x16x4_f32(
                false, a, false, b, (short)0, acc[j], false, false);
        }
    }

    // scale rows by dis[row] and write both hw (gather source) and agg (self-loop init)
    float dfac[8];
    #pragma unroll
    for (int v = 0; v < 8; ++v) dfac[v] = dis[m0 + v + 8 * hi];
    #pragma unroll
    for (int j = 0; j < 8; ++j) {
        #pragma unroll
        for (int v = 0; v < 8; ++v) {
            float val  = acc[j][v] * dfac[v];
            size_t idx = (size_t)(m0 + v + 8 * hi) * GH + j * 16 + nl;
            hw[idx]  = val;
            agg[idx] = val;
        }
    }
}

// ---------------- edge scatter: agg[col] += hw[row]  (norm hoisted out) --------------
// 32 lanes per edge, float4 per lane -> coalesced 512B gather + 4 f32 atomics per lane.
// Both buffers (2 x 51 MB) are L2-resident on MI455X (192 MB L2).
__global__ __launch_bounds__(256) void k_scatter(const int* __restrict__ ei,
                          const float* __restrict__ src, float* __restrict__ agg)
{
    long long t = (long long)blockIdx.x * blockDim.x + threadIdx.x;
    long long e = t >> 5;
    int lane = (int)(t & 31);
    if (e >= GE) return;
    int r = ei[e];
    int c = ei[GE + e];
    const float4 v = *(const float4*)(src + (size_t)r * GH + lane * 4);
    float* dst = agg + (size_t)c * GH + lane * 4;
    atomicAdd(dst + 0, v.x);
    atomicAdd(dst + 1, v.y);
    atomicAdd(dst + 2, v.z);
    atomicAdd(dst + 3, v.w);
}

// ---------------- finalize: out = agg * dis[col] + bias ----------------
__global__ void k_finalize(const float* __restrict__ agg, const float* __restrict__ dis,
                           const float* __restrict__ bias, float* __restrict__ out)
{
    long long i = (long long)blockIdx.x * blockDim.x + threadIdx.x;
    if (i >= (long long)GN * GH) return;
    int n = (int)(i >> 7);
    int f = (int)(i & 127);
    out[i] = agg[i] * dis[n] + bias[f];
}

// ---------------- batchnorm (training stats) + relu ----------------
__global__ void k_zero256(float* __restrict__ p) {
    p[threadIdx.x] = 0.0f;                           // sum[128] + sumsq[128]
}

__global__ __launch_bounds__(128) void k_bn_stats(const float* __restrict__ z,
                                                  float* __restrict__ sum,
                                                  float* __restrict__ sumsq)
{
    int f = threadIdx.x;                             // one feature per thread (coalesced)
    float s = 0.0f, q = 0.0f;
    for (int r = blockIdx.x; r < GN; r += gridDim.x) {
        float v = z[(size_t)r * GH + f];
        s += v; q += v * v;
    }
    atomicAdd(&sum[f], s);
    atomicAdd(&sumsq[f], q);
}

__global__ void k_bn_params(const float* __restrict__ sum, const float* __restrict__ sumsq,
                            const float* __restrict__ g, const float* __restrict__ be,
                            float* __restrict__ scale, float* __restrict__ shift)
{
    int f = threadIdx.x;
    float invn = 1.0f / (float)GN;
    float mu  = sum[f] * invn;
    float var = sumsq[f] * invn - mu * mu;           // biased var, torch BN training
    float sc  = g[f] * rsqrtf(var + GEPS);
    scale[f] = sc;
    shift[f] = be[f] - mu * sc;
}

__global__ void k_bn_apply_relu(float* __restrict__ z, const float* __restrict__ scale,
                                const float* __restrict__ shift)
{
    long long i = (long long)blockIdx.x * blockDim.x + threadIdx.x;
    if (i >= (long long)GN * GH) return;
    int f = (int)(i & 127);
    z[i] = fmaxf(z[i] * scale[f] + shift[f], 0.0f);
}

// ---------------- host orchestration ----------------
extern "C" void kernel_launch(void* const* d_in, const int* in_sizes, int n_in,
                              void* d_out, int out_size, void* d_ws, size_t ws_size,
                              hipStream_t stream) {
    const float* x   = (const float*)d_in[0];
    const int*   ei  = (const int*)  d_in[1];   // [2][E] row-major
    const float* W1  = (const float*)d_in[2];
    const float* b1  = (const float*)d_in[3];
    const float* W2  = (const float*)d_in[4];
    const float* b2  = (const float*)d_in[5];
    const float* W3  = (const float*)d_in[6];
    const float* b3  = (const float*)d_in[7];
    const float* g1  = (const float*)d_in[8];
    const float* be1 = (const float*)d_in[9];
    const float* g2  = (const float*)d_in[10];
    const float* be2 = (const float*)d_in[11];

    float* ws    = (float*)d_ws;
    float* dis   = ws;                     // [N]
    float* stats = ws + GN;                // sum[128] sumsq[128] scale[128] shift[128]
    float* bufA  = ws + GN + 512;          // [N*H]  (also deg scratch before layer 1)
    float* bufB  = bufA + (size_t)GN * GH; // [N*H]
    float* bufC  = bufB + (size_t)GN * GH; // [N*H]
    float* sum   = stats, *sumsq = stats + 128, *scale = stats + 256, *shift = stats + 384;

    const int T = 256;
    const int gN    = (GN + T - 1) / T;
    const int gE    = (GE + T - 1) / T;
    const int gNH   = (GN * GH + T - 1) / T;     // 50000
    const int gScat = (int)(((long long)GE * 32 + T - 1) / T); // 200000
    const int gGemm = (GN / 16 + 3) / 4;         // 4 waves (strips) per block

    // --- symmetric normalization ---
    k_deg_init <<<gN, T, 0, stream>>>(bufA);
    k_deg_edges<<<gE, T, 0, stream>>>(ei, bufA);
    k_dis      <<<gN, T, 0, stream>>>(bufA, dis);

    // --- layer 1: x -> bufB ---
    k_gemm_wmma<<<gGemm, 128, 0, stream>>>(x, W1, dis, bufA, bufB);
    k_scatter  <<<gScat, T, 0, stream>>>(ei, bufA, bufB);
    k_finalize <<<gNH, T, 0, stream>>>(bufB, dis, b1, bufB);
    k_zero256  <<<1, 256, 0, stream>>>(stats);
    k_bn_stats <<<128, 128, 0, stream>>>(bufB, sum, sumsq);
    k_bn_params<<<1, 128, 0, stream>>>(sum, sumsq, g1, be1, scale, shift);
    k_bn_apply_relu<<<gNH, T, 0, stream>>>(bufB, scale, shift);

    // --- layer 2: bufB -> bufC ---
    k_gemm_wmma<<<gGemm, 128, 0, stream>>>(bufB, W2, dis, bufA, bufC);
    k_scatter  <<<gScat, T, 0, stream>>>(ei, bufA, bufC);
    k_finalize <<<gNH, T, 0, stream>>>(bufC, dis, b2, bufC);
    k_zero256  <<<1, 256, 0, stream>>>(stats);
    k_bn_stats <<<128, 128, 0, stream>>>(bufC, sum, sumsq);
    k_bn_params<<<1, 128, 0, stream>>>(sum, sumsq, g2, be2, scale, shift);
    k_bn_apply_relu<<<gNH, T, 0, stream>>>(bufC, scale, shift);

    // --- layer 3: bufC -> d_out ---
    k_gemm_wmma<<<gGemm, 128, 0, stream>>>(bufC, W3, dis, bufA, bufB);
    k_scatter  <<<gScat, T, 0, stream>>>(ei, bufA, bufB);
    k_finalize <<<gNH, T, 0, stream>>>(bufB, dis, b3, (float*)d_out);
}